// TensorSnake_3350074491131
// MI455X (gfx1250) — compile-verified
//
#include <hip/hip_runtime.h>
#include <cstdint>

// TensorSnake step for MI455X (gfx1250, wave32).
// Design: 1 wave == 1 game (65536 games). 8 waves / 256-thread block.
// Memory-bound: ~512-768 MB per call -> ~22-33 us at 23.3 TB/s HBM.
//  - b128 NT loads/stores (stream > L2 capacity, avoid cache thrash)
//  - wave-uniform early exits: dead games skip the board read entirely,
//    non-feeding games skip the 256 MB noise read entirely
//  - argmax (first-max, jnp.argmax semantics) via 5-step shfl_xor in-wave
//  - CDNA5 async path: noise staged global->LDS with
//    global_load_async_to_lds_b128 + s_wait_asynccnt (wave-private LDS,
//    so no workgroup barrier needed; ASYNCcnt is per-wave)

#define BOARD 32
#define CELLS (BOARD * BOARD)
#define WAVES_PER_BLOCK 8
#define BLOCK_THREADS (WAVES_PER_BLOCK * 32)

typedef int   vi4 __attribute__((ext_vector_type(4)));
typedef float vf4 __attribute__((ext_vector_type(4)));

#if defined(__gfx1250__) && __has_builtin(__builtin_amdgcn_global_load_async_to_lds_b128)
#define HAVE_ASYNC_LDS 1
// Builtin prototype (from hipcc diagnostic): param0 is
// '__attribute__((__vector_size__(4*sizeof(int)))) int __device__ *'
// i.e. pointer-to-v4i32 in AS(1); LDS side is the AS(3) counterpart.
typedef int b128_t __attribute__((vector_size(16)));
typedef __attribute__((address_space(1))) b128_t gb128_t;
typedef __attribute__((address_space(3))) b128_t lb128_t;
#else
#define HAVE_ASYNC_LDS 0
#endif

__global__ __launch_bounds__(BLOCK_THREADS) void snake_step_kernel(
    const float* __restrict__ noise,
    const int*   __restrict__ action,
    const int*   __restrict__ state,
    const int*   __restrict__ pos_cur,
    const int*   __restrict__ pos_last,
    const int*   __restrict__ pos_food,
    float*       __restrict__ out,
    int games)
{
#if HAVE_ASYNC_LDS
    __shared__ __align__(16) float lnoise[WAVES_PER_BLOCK][CELLS]; // 32 KB / block
#endif
    const int lane = threadIdx.x & 31;
    const int wib  = threadIdx.x >> 5;
    const int g    = blockIdx.x * WAVES_PER_BLOCK + wib;
    if (g >= games) return;

    const int*   S = state + (size_t)g * CELLS;
    const float* N = noise + (size_t)g * CELLS;
    float*       O = out   + (size_t)g * CELLS;

    // ---- wave-uniform scalar game logic (all lanes redundantly; loads coalesce) ----
    const int act = action[g];
    const int cx = pos_cur[2*g+0],  cy = pos_cur[2*g+1];
    const int lx = pos_last[2*g+0], ly = pos_last[2*g+1];
    const int fx = pos_food[2*g+0], fy = pos_food[2*g+1];
    const int curIdx  = cx * BOARD + cy;
    const int lastIdx = lx * BOARD + ly;

    int ad = S[curIdx];
    ad += (act == 2) - (act == 0);
    ad = (ad == 1) ? 5 : (ad == 6) ? 2 : ad;
    // DIRS: 2:(-1,0) 3:(0,1) 4:(1,0) 5:(0,-1); others (0,0)
    const int dx = (ad == 2) ? -1 : (ad == 4) ? 1 : 0;
    const int dy = (ad == 3) ?  1 : (ad == 5) ? -1 : 0;
    const int nx = (cx + dx) & (BOARD - 1);
    const int ny = (cy + dy) & (BOARD - 1);
    const int nextIdx = nx * BOARD + ny;
    const bool dead    = S[nextIdx] > 1;          // next != cur always (|dir|=1 mod 32)
    const bool feeding = (nx == fx) && (ny == fy);

    // ---- dead: emit init board, skip all bulk reads (saves 8 KB/game) ----
    if (dead) {
        #pragma unroll
        for (int j = 0; j < 8; ++j) {
            const int base = j * 128 + lane * 4;
            vf4 o;
            #pragma unroll
            for (int e = 0; e < 4; ++e) {
                const int i = base + e;
                o[e] = (i == 0 || i == 1) ? 3.0f : (i == 2) ? 1.0f : 0.0f;
            }
            __builtin_nontemporal_store(o, (vf4*)(O + base));
        }
        return;
    }

#if HAVE_ASYNC_LDS
    // ---- feeding path: kick off async noise staging global -> LDS ----
    if (feeding) {
        const uint32_t lbase = (uint32_t)(uintptr_t)&lnoise[wib][lane * 4];
        #pragma unroll
        for (int j = 0; j < 8; ++j) {
            __builtin_amdgcn_global_load_async_to_lds_b128(
                (gb128_t*)(uintptr_t)(N + j * 128 + lane * 4),
                (lb128_t*)(uintptr_t)(lbase + j * 512),
                /*offset=*/0, /*cpol=*/0);
        }
    }
#endif

    // ---- stream the board: 8 x b128 NT loads per lane (coalesced 512B/instr) ----
    vi4 sv[8];
    #pragma unroll
    for (int j = 0; j < 8; ++j)
        sv[j] = __builtin_nontemporal_load((const vi4*)(S + j * 128 + lane * 4));

    // ---- argmax over (empty ? noise : -1): only consequential when feeding ----
    int foodIdx = -1;
    if (feeding) {
#if HAVE_ASYNC_LDS
#if __has_builtin(__builtin_amdgcn_s_wait_asynccnt)
        __builtin_amdgcn_s_wait_asynccnt(0);
#else
        asm volatile("s_wait_asynccnt 0" ::: "memory");
#endif
        asm volatile("" ::: "memory");
#endif
        float bestV = -1.0f;   // all keys >= -1; all -1 -> idx 0 (jnp.argmax)
        int   bestI = 0;
        #pragma unroll
        for (int j = 0; j < 8; ++j) {
            const int base = j * 128 + lane * 4;
#if HAVE_ASYNC_LDS
            vf4 nv = *(const vf4*)&lnoise[wib][base];
#else
            vf4 nv = __builtin_nontemporal_load((const vf4*)(N + base));
#endif
            vi4 s = sv[j];
            #pragma unroll
            for (int e = 0; e < 4; ++e) {
                const int i = base + e;
                int v = s[e];
                if (i == curIdx) v = ad;       // scan sees post-head-turn board;
                                               // tail override is identity when feeding
                const float key = (v == 0) ? nv[e] : -1.0f;
                if (key > bestV) { bestV = key; bestI = i; }  // first-max in lane
            }
        }
        // cross-lane first-max: larger value, then smaller index (wave32)
        #pragma unroll
        for (int off = 16; off > 0; off >>= 1) {
            const float ov = __shfl_xor(bestV, off, 32);
            const int   oi = __shfl_xor(bestI, off, 32);
            if (ov > bestV || (ov == bestV && oi < bestI)) { bestV = ov; bestI = oi; }
        }
        foodIdx = bestI;
    }

    // ---- emit: reference write order cur -> last -> food -> next ----
    #pragma unroll
    for (int j = 0; j < 8; ++j) {
        const int base = j * 128 + lane * 4;
        vi4 s = sv[j];
        vf4 o;
        #pragma unroll
        for (int e = 0; e < 4; ++e) {
            const int i = base + e;
            int v = s[e];
            if (i == curIdx) v = ad;
            if (!feeding && i == lastIdx) v = 0;      // feeding keeps old_tail (== v)
            if (feeding && i == foodIdx) v = 1;
            if (i == nextIdx) v = ad;
            o[e] = (float)v;
        }
        __builtin_nontemporal_store(o, (vf4*)(O + base));
    }
}

extern "C" void kernel_launch(void* const* d_in, const int* in_sizes, int n_in,
                              void* d_out, int out_size, void* d_ws, size_t ws_size,
                              hipStream_t stream) {
    (void)n_in; (void)out_size; (void)d_ws; (void)ws_size;
    const float* noise    = (const float*)d_in[0];
    const int*   action   = (const int*)  d_in[1];
    const int*   state    = (const int*)  d_in[2];
    const int*   pos_cur  = (const int*)  d_in[3];
    const int*   pos_last = (const int*)  d_in[4];
    const int*   pos_food = (const int*)  d_in[5];
    float*       out      = (float*)      d_out;

    const int games = in_sizes[1];  // action has one entry per game
    const int blocks = (games + WAVES_PER_BLOCK - 1) / WAVES_PER_BLOCK;
    snake_step_kernel<<<blocks, BLOCK_THREADS, 0, stream>>>(
        noise, action, state, pos_cur, pos_last, pos_food, out, games);
}